// TrainableLDAHead_40793599377495
// MI455X (gfx1250) — compile-verified
//
#include <hip/hip_runtime.h>
#include <hip/hip_bf16.h>
#include <math.h>

// Problem constants (reference: B=1024, C=1000, D=256, EPS=1e-4)
#define B_N 1024
#define C_N 1000
#define D_N 256
#define EPS_V 1e-4f

typedef __attribute__((ext_vector_type(2))) float v2f;
typedef __attribute__((ext_vector_type(8))) float v8f;

// One 16x16x4 f32 WMMA step: D = A(16x4) * B(4x16) + C
static __device__ __forceinline__ v8f wmma_step(v2f a, v2f b, v8f c) {
  // 8 args: (neg_a, A, neg_b, B, c_mod, C, reuse_a, reuse_b)
  return __builtin_amdgcn_wmma_f32_16x16x4_f32(false, a, false, b, (short)0, c,
                                               false, false);
}

// -------------------------------------------------------------------------
// P = F * F^T + EPS*I   (F is DxD row-major; FF^T is exactly symmetric, so
// the reference's 0.5*(P+P^T) symmetrization is a no-op)
// grid = (D/16, D/16), block = 32 (one wave per 16x16 tile)
// -------------------------------------------------------------------------
__global__ __launch_bounds__(32) void lda_precision(const float* __restrict__ F,
                                                    float* __restrict__ P) {
  const int lane = threadIdx.x;
  const int lo = lane & 15, hi = lane >> 4;
  const int i0 = blockIdx.x * 16, j0 = blockIdx.y * 16;
  // A[m][k] = F[i0+m][k]   -> lane loads F[i0+lo][k0+2*hi +{0,1}]  (contig f2)
  // B[k][n] = F[j0+n][k]   -> lane loads F[j0+lo][k0+2*hi +{0,1}]  (contig f2)
  const float* arow = F + (i0 + lo) * D_N + 2 * hi;
  const float* brow = F + (j0 + lo) * D_N + 2 * hi;
  v8f acc = {};
#pragma unroll 8
  for (int k0 = 0; k0 < D_N; k0 += 4) {
    v2f a = *(const v2f*)(arow + k0);
    v2f b = *(const v2f*)(brow + k0);
    acc = wmma_step(a, b, acc);
  }
#pragma unroll
  for (int r = 0; r < 8; ++r) {
    const int i = i0 + r + 8 * hi;
    const int j = j0 + lo;
    float v = acc[r];
    if (i == j) v += EPS_V;
    P[i * D_N + j] = v;
  }
}

// -------------------------------------------------------------------------
// Y = X * P   (X is rows x D row-major, P is D x D row-major, symmetric)
// 32x16 output tile per wave: 2 A-fragments share each B-fragment.
// grid = (ceil(rows/32), D/16), block = 32
// -------------------------------------------------------------------------
__global__ __launch_bounds__(32) void lda_gemm_xP(const float* __restrict__ X,
                                                  const float* __restrict__ P,
                                                  float* __restrict__ Y,
                                                  int rows) {
  const int lane = threadIdx.x;
  const int lo = lane & 15, hi = lane >> 4;
  const int r0 = blockIdx.x * 32, d0 = blockIdx.y * 16;
  int ra0 = r0 + lo;       if (ra0 >= rows) ra0 = rows - 1;  // clamp loads
  int ra1 = r0 + 16 + lo;  if (ra1 >= rows) ra1 = rows - 1;
  const float* arow0 = X + ra0 * D_N + 2 * hi;
  const float* arow1 = X + ra1 * D_N + 2 * hi;
  const float* bcol = P + d0 + lo;  // + k*D_N walks down a column of P
  v8f acc0 = {}, acc1 = {};
#pragma unroll 4
  for (int k0 = 0; k0 < D_N; k0 += 4) {
    v2f b;
    b.x = bcol[(k0 + 2 * hi + 0) * D_N];
    b.y = bcol[(k0 + 2 * hi + 1) * D_N];
    v2f a0 = *(const v2f*)(arow0 + k0);
    v2f a1 = *(const v2f*)(arow1 + k0);
    acc0 = wmma_step(a0, b, acc0);
    acc1 = wmma_step(a1, b, acc1);
  }
#pragma unroll
  for (int r = 0; r < 8; ++r) {
    const int row0 = r0 + r + 8 * hi;
    const int row1 = row0 + 16;
    if (row0 < rows) Y[row0 * D_N + d0 + lo] = acc0[r];
    if (row1 < rows) Y[row1 * D_N + d0 + lo] = acc1[r];
  }
}

// -------------------------------------------------------------------------
// out[row] = dot(x[row,:], y[row,:])  -- one wave per row, wave32 reduction
// block = 128 (4 waves)
// -------------------------------------------------------------------------
__global__ void lda_rowdot(const float* __restrict__ x,
                           const float* __restrict__ y,
                           float* __restrict__ out, int rows) {
  const int wave = (int)((blockIdx.x * blockDim.x + threadIdx.x) >> 5);
  const int lane = threadIdx.x & 31;
  if (wave >= rows) return;
  const float* xr = x + wave * D_N;
  const float* yr = y + wave * D_N;
  float s = 0.f;
#pragma unroll
  for (int t = lane; t < D_N; t += 32) s += xr[t] * yr[t];
#pragma unroll
  for (int off = 16; off > 0; off >>= 1) s += __shfl_xor(s, off, 32);
  if (lane == 0) out[wave] = s;
}

// -------------------------------------------------------------------------
// log_prior = log_softmax(logits)  -- single block of 1024 threads
// -------------------------------------------------------------------------
__global__ void lda_logsoftmax(const float* __restrict__ logits,
                               float* __restrict__ out, int n) {
  __shared__ float red[1024];
  const int t = threadIdx.x;
  const float x = (t < n) ? logits[t] : -INFINITY;
  red[t] = x;
  __syncthreads();
  for (int off = 512; off > 0; off >>= 1) {
    if (t < off) red[t] = fmaxf(red[t], red[t + off]);
    __syncthreads();
  }
  const float m = red[0];
  __syncthreads();
  red[t] = (t < n) ? expf(x - m) : 0.f;
  __syncthreads();
  for (int off = 512; off > 0; off >>= 1) {
    if (t < off) red[t] += red[t + off];
    __syncthreads();
  }
  const float lse = m + logf(red[0]);
  if (t < n) out[t] = x - lse;
}

// -------------------------------------------------------------------------
// out[b,c] = logp[c] - 0.5*zPz[b] + dot(Pz[b,:], mu[c,:]) - 0.5*muPmu[c]
// 32x32 output tile per wave (2x2 register blocking): per K-step,
// 4 float2 loads feed 4 WMMAs -> 1 load per WMMA.
// grid = (B/32, ceil(C/32)), block = 32
// -------------------------------------------------------------------------
__global__ __launch_bounds__(32) void lda_final(
    const float* __restrict__ Pz, const float* __restrict__ mu,
    const float* __restrict__ zPz, const float* __restrict__ muPmu,
    const float* __restrict__ logp, float* __restrict__ out) {
  const int lane = threadIdx.x;
  const int lo = lane & 15, hi = lane >> 4;
  const int b0 = blockIdx.x * 32, c0 = blockIdx.y * 32;
  const float* arow0 = Pz + (b0 + lo) * D_N + 2 * hi;
  const float* arow1 = Pz + (b0 + 16 + lo) * D_N + 2 * hi;
  int cr0 = c0 + lo;       if (cr0 >= C_N) cr0 = C_N - 1;  // clamp loads
  int cr1 = c0 + 16 + lo;  if (cr1 >= C_N) cr1 = C_N - 1;
  const float* brow0 = mu + cr0 * D_N + 2 * hi;
  const float* brow1 = mu + cr1 * D_N + 2 * hi;
  v8f acc00 = {}, acc01 = {}, acc10 = {}, acc11 = {};
#pragma unroll 4
  for (int k0 = 0; k0 < D_N; k0 += 4) {
    v2f a0 = *(const v2f*)(arow0 + k0);
    v2f a1 = *(const v2f*)(arow1 + k0);
    v2f bb0 = *(const v2f*)(brow0 + k0);
    v2f bb1 = *(const v2f*)(brow1 + k0);
    acc00 = wmma_step(a0, bb0, acc00);
    acc01 = wmma_step(a0, bb1, acc01);
    acc10 = wmma_step(a1, bb0, acc10);
    acc11 = wmma_step(a1, bb1, acc11);
  }
  const int cA = c0 + lo;
  const int cB = c0 + 16 + lo;
  const float ctermA = (cA < C_N) ? (logp[cA] - 0.5f * muPmu[cA]) : 0.f;
  const float ctermB = (cB < C_N) ? (logp[cB] - 0.5f * muPmu[cB]) : 0.f;
#pragma unroll
  for (int r = 0; r < 8; ++r) {
    const int bA = b0 + r + 8 * hi;   // always < B (B is a multiple of 32)
    const int bB = bA + 16;
    const float zA = -0.5f * zPz[bA];
    const float zB = -0.5f * zPz[bB];
    if (cA < C_N) {
      out[bA * C_N + cA] = ctermA + acc00[r] + zA;
      out[bB * C_N + cA] = ctermA + acc10[r] + zB;
    }
    if (cB < C_N) {
      out[bA * C_N + cB] = ctermB + acc01[r] + zA;
      out[bB * C_N + cB] = ctermB + acc11[r] + zB;
    }
  }
}

extern "C" void kernel_launch(void* const* d_in, const int* in_sizes, int n_in,
                              void* d_out, int out_size, void* d_ws,
                              size_t ws_size, hipStream_t stream) {
  (void)in_sizes; (void)n_in; (void)out_size; (void)ws_size;
  const float* z      = (const float*)d_in[0];  // (B, D)
  const float* mu     = (const float*)d_in[1];  // (C, D)
  const float* logits = (const float*)d_in[2];  // (C,)
  const float* F      = (const float*)d_in[3];  // (D, D)
  float* out = (float*)d_out;                   // (B, C)

  // Workspace layout (floats): P | Pz | muP | zPz | muPmu | logp  (~2.3 MB)
  float* w     = (float*)d_ws;
  float* P     = w;                 // D*D      = 65536
  float* Pz    = P + D_N * D_N;     // B*D      = 262144
  float* muP   = Pz + B_N * D_N;    // C*D      = 256000
  float* zPz   = muP + C_N * D_N;   // B        = 1024
  float* muPmu = zPz + B_N;         // C        = 1000
  float* logp  = muPmu + C_N;       // C        = 1000

  lda_precision<<<dim3(D_N / 16, D_N / 16), 32, 0, stream>>>(F, P);
  lda_gemm_xP<<<dim3(B_N / 32, D_N / 16), 32, 0, stream>>>(z, P, Pz, B_N);
  lda_gemm_xP<<<dim3((C_N + 31) / 32, D_N / 16), 32, 0, stream>>>(mu, P, muP,
                                                                  C_N);
  lda_rowdot<<<dim3((B_N + 3) / 4), 128, 0, stream>>>(z, Pz, zPz, B_N);
  lda_rowdot<<<dim3((C_N + 3) / 4), 128, 0, stream>>>(mu, muP, muPmu, C_N);
  lda_logsoftmax<<<1, 1024, 0, stream>>>(logits, logp, C_N);
  lda_final<<<dim3(B_N / 32, (C_N + 31) / 32), 32, 0, stream>>>(
      Pz, mu, zPz, muPmu, logp, out);
}